// MaxPooling_64888365908328
// MI455X (gfx1250) — compile-verified
//
#include <hip/hip_runtime.h>
#include <hip/hip_bf16.h>
#include <stdint.h>

// ---------------- static problem config (mirrors reference) ----------------
#define NN      500000            // nodes
#define FF      64                // feature dim
#define EE      4000000           // edges
#define BBATCH  16
#define NXX     41
#define NYY     41
#define NMAX    26896             // BBATCH*NXX*NYY
#define SENTK   723394816u        // NMAX*NMAX (fits int32)
#define WORDS_E 22606088          // SENTK/32  (edge-key bitmap words, 90.4 MB -> L2 resident)
#define NBLK_E  11039             // ceil(WORDS_E / 2048)
#define NTILES  7813              // ceil(NN / 64)
#define VBM_W   841               // ceil(NMAX/32)

// d_out float layout (concatenated reference outputs)
#define O_X     0                 // [NMAX,64]
#define O_POS   1721344           // [NMAX,3]
#define O_BATCH 1802032           // [NMAX]
#define O_EI    1828928           // [2,EE]
#define O_EA    9828928           // [EE,3]

// workspace layout (32-bit words)
#define WS_VOX    0
#define WS_INV    500000
#define WS_CNT    1000000
#define WS_FIRST  1026896
#define WS_PSUM   1053792
#define WS_XKEY   1134480
#define WS_VBM    2855824
#define WS_VRANK  2856665
#define WS_MIN    2883561         // 2 words
#define WS_NUNIQ  2883563
#define WS_MAXABS 2883564
#define WS_BSUM   2883565
#define WS_BOFF   2894605
#define WS_EBM    2905645         // 22,606,088 words

// native clang vector (accepted by __builtin_nontemporal_store, unlike HIP's uint4)
typedef unsigned int v4u __attribute__((ext_vector_type(4)));

// order-preserving float<->uint key (for atomicMax-based segment max)
__device__ __forceinline__ unsigned f2key(float f) {
  unsigned u = __float_as_uint(f);
  return (u & 0x80000000u) ? ~u : (u | 0x80000000u);
}
__device__ __forceinline__ float key2f(unsigned k) {
  unsigned u = (k & 0x80000000u) ? (k & 0x7fffffffu) : ~k;
  return __uint_as_float(u);
}

// gfx1250 async global->LDS DMA (ASYNCcnt path, CDNA5 ISA)
__device__ __forceinline__ void async_load_b128(unsigned lds_byte_addr, const void* gaddr) {
  asm volatile("global_load_async_to_lds_b128 %0, %1, off"
               :: "v"(lds_byte_addr), "v"((unsigned long long)(size_t)gaddr)
               : "memory");
}
__device__ __forceinline__ void wait_asynccnt0() {
  asm volatile("s_wait_asynccnt 0" ::: "memory");
}
// partial wait: older async ops complete in order, so <=4 outstanding means the
// previous tile's 4 DMA chunks have landed while the next tile's 4 stay in flight.
__device__ __forceinline__ void wait_asynccnt4() {
  asm volatile("s_wait_asynccnt 4" ::: "memory");
}

// ---------------- init ----------------
__global__ void k_init_small(int* cnt, int* first, float* psum, unsigned* xkeys,
                             unsigned* vbm, int* minxy, int* nuniq, int* maxabs) {
  int stride = gridDim.x * blockDim.x;
  for (int idx = blockIdx.x * blockDim.x + threadIdx.x; idx < NMAX * FF; idx += stride) {
    xkeys[idx] = 0u;                     // key(any real float) > 0, so 0 == "empty"
    if (idx < NMAX)     { cnt[idx] = 0; first[idx] = 0x7fffffff; }
    if (idx < 3 * NMAX) psum[idx] = 0.0f;
    if (idx < VBM_W)    vbm[idx] = 0u;
    if (idx == 0) { minxy[0] = 0x7f800000; minxy[1] = 0x7f800000; *nuniq = 0; *maxabs = 0; }
  }
}
// b128 non-temporal zeroing of the 90.4 MB bitmap (WORDS_E % 4 == 0)
__global__ void k_zero_ebm(v4u* __restrict__ ebm4) {
  int stride = gridDim.x * blockDim.x;
  v4u z = {0u, 0u, 0u, 0u};
  for (int i = blockIdx.x * blockDim.x + threadIdx.x; i < WORDS_E / 4; i += stride)
    __builtin_nontemporal_store(z, &ebm4[i]);
}

// ---------------- 1) min over pos[:, :2]  (pos >= 0 -> int order == float order) ----------------
__global__ void k_minpos(const float* __restrict__ pos, int* minxy) {
  __shared__ int sx[256], sy[256];
  int t = threadIdx.x;
  int mx = 0x7f800000, my = 0x7f800000;
  for (int i = blockIdx.x * blockDim.x + t; i < NN; i += gridDim.x * blockDim.x) {
    int px = __float_as_int(pos[i * 3 + 0]);
    int py = __float_as_int(pos[i * 3 + 1]);
    mx = px < mx ? px : mx;  my = py < my ? py : my;
  }
  sx[t] = mx; sy[t] = my; __syncthreads();
  for (int o = 128; o > 0; o >>= 1) {
    if (t < o) { sx[t] = min(sx[t], sx[t + o]); sy[t] = min(sy[t], sy[t + o]); }
    __syncthreads();
  }
  if (t == 0) { atomicMin(&minxy[0], sx[0]); atomicMin(&minxy[1], sy[0]); }
}

// ---------------- 2) voxel id per node + voxel-present bitmap ----------------
__global__ void k_vox(const float* __restrict__ pos, const int* __restrict__ batch,
                      const int* minxy, int* __restrict__ vox, unsigned* __restrict__ vbm) {
  float sx = __int_as_float(minxy[0]), sy = __int_as_float(minxy[1]);
  for (int i = blockIdx.x * blockDim.x + threadIdx.x; i < NN; i += gridDim.x * blockDim.x) {
    int cx = (int)floorf((pos[i * 3 + 0] - sx) / 0.025f);
    int cy = (int)floorf((pos[i * 3 + 1] - sy) / 0.025f);
    cx = min(max(cx, 0), NXX - 1);
    cy = min(max(cy, 0), NYY - 1);
    int v = batch[i] * (NXX * NYY) + cy * NXX + cx;
    vox[i] = v;
    atomicOr(&vbm[v >> 5], 1u << (v & 31));
  }
}

// ---------------- 3) rank occupied voxels -> consecutive cluster ids (sorted) ----------------
__global__ void k_rank(const unsigned* __restrict__ vbm, int* __restrict__ vrank) {
  __shared__ int s[1024];
  int t = threadIdx.x;
  unsigned w = (t < VBM_W) ? vbm[t] : 0u;
  int pc = __popc(w);
  s[t] = pc; __syncthreads();
  for (int off = 1; off < 1024; off <<= 1) {
    int v = (t >= off) ? s[t - off] : 0;
    __syncthreads();
    s[t] += v;
    __syncthreads();
  }
  int r = s[t] - pc;   // exclusive prefix
  if (t < VBM_W) {
    for (int b = 0; b < 32; ++b) {
      int v = t * 32 + b;
      if (v < NMAX) vrank[v] = ((w >> b) & 1u) ? r++ : -1;
    }
  }
}

// ---------------- 4) pooled scatter: double-buffered async-LDS DMA pipeline ----------------
__global__ void k_pool(const float* __restrict__ x, const float* __restrict__ pos,
                       const int* __restrict__ vox, const int* __restrict__ vrank,
                       int* __restrict__ inv, int* __restrict__ cnt, int* __restrict__ first,
                       float* __restrict__ psum, unsigned* __restrict__ xkeys) {
  __shared__ __align__(16) float sbuf[2][64 * FF];   // 2 x 16 KB tile of x
  __shared__ int scid[2][64];
  const int t = threadIdx.x;                         // 256 threads = 8 waves

  auto issue = [&](int tile, int b) {
    if (tile >= NTILES) return;
    const int base = tile * 64;
    unsigned ldsbase = (unsigned)(size_t)(void*)sbuf[b];
#pragma unroll
    for (int k = 0; k < 4; ++k) {
      int q = t + k * 256;                           // 1024 x 16B chunks per tile
      long long ge = (long long)base * FF + (long long)q * 4;
      long long gmax = (long long)NN * FF - 4;
      if (ge > gmax) ge = gmax;                      // clamp tail (masked via scid)
      async_load_b128(ldsbase + (unsigned)q * 16u, x + ge);
    }
  };

  int buf = 0;
  issue(blockIdx.x, 0);                              // prime the pipeline
  for (int tile = blockIdx.x; tile < NTILES; tile += gridDim.x, buf ^= 1) {
    const int next = tile + gridDim.x;
    // start next tile's DMA into the other buffer (fully consumed at the
    // barrier that ended the previous iteration -> no WAR hazard)
    issue(next, buf ^ 1);
    // cluster id + scalar pooling for this tile's 64 nodes (overlaps DMA)
    if (t < 64) {
      int i = tile * 64 + t;
      int c = -1;
      if (i < NN) {
        c = vrank[vox[i]];
        inv[i] = c;
        atomicAdd(&cnt[c], 1);
        atomicMin(&first[c], i);
        atomicAdd(&psum[c * 3 + 0], pos[i * 3 + 0]);
        atomicAdd(&psum[c * 3 + 1], pos[i * 3 + 1]);
        atomicAdd(&psum[c * 3 + 2], pos[i * 3 + 2]);
      }
      scid[buf][t] = c;
    }
    if (next < NTILES) wait_asynccnt4();             // current tile landed, next in flight
    else               wait_asynccnt0();             // drain at pipeline tail
    __syncthreads();                                 // all waves' DMA + scid visible
    // segment-max of features: 64 nodes x 64 feats over 256 lanes
    const int f = t & 63;
    const int r0 = t >> 6;
#pragma unroll 4
    for (int m = 0; m < 16; ++m) {
      int node = m * 4 + r0;
      int c = scid[buf][node];
      if (c >= 0) atomicMax(&xkeys[c * FF + f], f2key(sbuf[buf][node * FF + f]));
    }
    __syncthreads();                                 // buffer free for refill next iter
  }
}

// ---------------- 5) finalize pooled outputs ----------------
__global__ void k_fin_x(const v4u* __restrict__ xkeys4, const int* __restrict__ cnt,
                        float* __restrict__ ox) {
  int idx = blockIdx.x * blockDim.x + threadIdx.x;   // NMAX*FF/4 quads
  if (idx < NMAX * FF / 4) {
    int c = idx >> 4;                                // 16 quads per cluster row
    v4u k = xkeys4[idx];
    float4 v;
    if (cnt[c] > 0) v = make_float4(key2f(k.x), key2f(k.y), key2f(k.z), key2f(k.w));
    else            v = make_float4(0.f, 0.f, 0.f, 0.f);
    ((float4*)ox)[idx] = v;
  }
}
__global__ void k_fin_pos(const int* __restrict__ cnt, const float* __restrict__ psum,
                          const int* __restrict__ first, const int* __restrict__ batch,
                          float* __restrict__ opos, float* __restrict__ obatch) {
  int c = blockIdx.x * blockDim.x + threadIdx.x;
  if (c < NMAX) {
    int n = cnt[c];
    float dv = (float)(n > 0 ? n : 1);
    opos[c * 3 + 0] = psum[c * 3 + 0] / dv;
    opos[c * 3 + 1] = psum[c * 3 + 1] / dv;
    opos[c * 3 + 2] = psum[c * 3 + 2] / dv;
    obatch[c] = (n > 0) ? (float)batch[first[c]] : -1.0f;
  }
}

// ---------------- 6) edge dedupe: bitmap mark (L2-resident, 90.4 MB < 192 MB) ----------------
__global__ void k_mark_edges(const int* __restrict__ e0, const int* __restrict__ e1,
                             const int* __restrict__ inv, unsigned* __restrict__ ebm) {
  for (int e = blockIdx.x * blockDim.x + threadIdx.x; e < EE; e += gridDim.x * blockDim.x) {
    __builtin_prefetch(e0 + e + 16384, 0, 0);   // global_prefetch_b8
    int r = inv[e0[e]];
    int c = inv[e1[e]];
    if (r != c) {
      unsigned key = (unsigned)r * (unsigned)NMAX + (unsigned)c;
      atomicOr(&ebm[key >> 5], 1u << (key & 31));
    }
  }
}

// ---------------- 7) ordered compaction of set bits -> sorted unique edges ----------------
__global__ void k_blockcount(const unsigned* __restrict__ ebm, int* __restrict__ bsum) {
  __shared__ int s[256];
  int t = threadIdx.x;
  int base = blockIdx.x * 2048 + t * 8;
  int cnt = 0;
#pragma unroll
  for (int k = 0; k < 8; ++k) {
    int w = base + k;
    if (w < WORDS_E) cnt += __popc(ebm[w]);
  }
  s[t] = cnt; __syncthreads();
  for (int o = 128; o > 0; o >>= 1) { if (t < o) s[t] += s[t + o]; __syncthreads(); }
  if (t == 0) bsum[blockIdx.x] = s[0];
}

__global__ void k_scan(const int* __restrict__ bsum, int* __restrict__ boff, int* nuniq) {
  __shared__ int s[1024];
  __shared__ int run;
  int t = threadIdx.x;
  if (t == 0) run = 0;
  __syncthreads();
  for (int start = 0; start < NBLK_E; start += 1024) {
    int i = start + t;
    int v = (i < NBLK_E) ? bsum[i] : 0;
    s[t] = v; __syncthreads();
    for (int off = 1; off < 1024; off <<= 1) {
      int a = (t >= off) ? s[t - off] : 0;
      __syncthreads();
      s[t] += a;
      __syncthreads();
    }
    if (i < NBLK_E) boff[i] = run + (s[t] - v);
    __syncthreads();
    int tot = s[1023];
    __syncthreads();
    if (t == 0) run += tot;
    __syncthreads();
  }
  if (t == 0) *nuniq = run;
}

__global__ void k_emit(const unsigned* __restrict__ ebm, const int* __restrict__ boff,
                       float* __restrict__ eif /* int staging in ei output region */) {
  __shared__ int s[256];
  int* out_i = (int*)eif;
  int t = threadIdx.x;
  int base = blockIdx.x * 2048 + t * 8;
  unsigned w[8];
  int cnt = 0;
#pragma unroll
  for (int k = 0; k < 8; ++k) {
    w[k] = (base + k < WORDS_E) ? ebm[base + k] : 0u;
    cnt += __popc(w[k]);
  }
  s[t] = cnt; __syncthreads();
  for (int off = 1; off < 256; off <<= 1) {
    int a = (t >= off) ? s[t - off] : 0;
    __syncthreads();
    s[t] += a;
    __syncthreads();
  }
  int pos = boff[blockIdx.x] + (s[t] - cnt);
#pragma unroll
  for (int k = 0; k < 8; ++k) {
    unsigned ww = w[k];
    unsigned keybase = ((unsigned)(base + k)) << 5;
    while (ww) {
      int b = __ffs(ww) - 1;
      ww &= ww - 1u;
      unsigned key = keybase + (unsigned)b;
      out_i[pos]      = (int)(key / (unsigned)NMAX);   // row
      out_i[EE + pos] = (int)(key % (unsigned)NMAX);   // col
      ++pos;
    }
  }
}

// ---------------- 8) cart = pos_pool[row]-pos_pool[col]; global max|.| ----------------
__global__ void k_cart(const float* __restrict__ pp, const float* __restrict__ eif,
                       float* __restrict__ ea, const int* nuniq, int* maxabs) {
  __shared__ int sm[256];
  const int* ei = (const int*)eif;
  int nu = *nuniq;
  int t = threadIdx.x;
  int lm = 0;
  for (int e = blockIdx.x * blockDim.x + t; e < nu; e += gridDim.x * blockDim.x) {
    int r = ei[e], c = ei[EE + e];
    float a0 = pp[r * 3 + 0] - pp[c * 3 + 0];
    float a1 = pp[r * 3 + 1] - pp[c * 3 + 1];
    float a2 = pp[r * 3 + 2] - pp[c * 3 + 2];
    ea[e * 3 + 0] = a0; ea[e * 3 + 1] = a1; ea[e * 3 + 2] = a2;
    int m = __float_as_int(fmaxf(fabsf(a0), fmaxf(fabsf(a1), fabsf(a2))));
    lm = m > lm ? m : lm;   // abs values >= 0 -> int order == float order
  }
  sm[t] = lm; __syncthreads();
  for (int o = 128; o > 0; o >>= 1) { if (t < o) sm[t] = max(sm[t], sm[t + o]); __syncthreads(); }
  if (t == 0) atomicMax(maxabs, sm[0]);
}

// ---------------- 9) finalize edges: in-place int->float + scale; pad tails ----------------
__global__ void k_fin_edges(float* __restrict__ eif, float* __restrict__ ea,
                            const int* nuniq, const int* maxabs) {
  int nu = *nuniq;
  float mv = __int_as_float(*maxabs);
  float scale = (mv > 0.0f) ? 0.5f / mv : 0.0f;
  int* ii = (int*)eif;
  for (int e = blockIdx.x * blockDim.x + threadIdx.x; e < EE; e += gridDim.x * blockDim.x) {
    if (e < nu) {
      int r = ii[e], c = ii[EE + e];          // read ints before overwriting as float
      eif[e] = (float)r;
      eif[EE + e] = (float)c;
      ea[e * 3 + 0] = ea[e * 3 + 0] * scale + 0.5f;
      ea[e * 3 + 1] = ea[e * 3 + 1] * scale + 0.5f;
      ea[e * 3 + 2] = ea[e * 3 + 2] * scale + 0.5f;
    } else {
      eif[e] = -1.0f; eif[EE + e] = -1.0f;
      ea[e * 3 + 0] = 0.0f; ea[e * 3 + 1] = 0.0f; ea[e * 3 + 2] = 0.0f;
    }
  }
}

extern "C" void kernel_launch(void* const* d_in, const int* in_sizes, int n_in,
                              void* d_out, int out_size, void* d_ws, size_t ws_size,
                              hipStream_t stream) {
  (void)in_sizes; (void)n_in; (void)out_size; (void)ws_size;
  const float* x     = (const float*)d_in[0];
  const float* pos   = (const float*)d_in[1];
  const int*   batch = (const int*)d_in[2];
  const int*   ei0   = (const int*)d_in[3];
  const int*   ei1   = ei0 + EE;

  int*      ws      = (int*)d_ws;
  int*      vox     = ws + WS_VOX;
  int*      inv     = ws + WS_INV;
  int*      cnt     = ws + WS_CNT;
  int*      first   = ws + WS_FIRST;
  float*    psum    = (float*)(ws + WS_PSUM);
  unsigned* xkeys   = (unsigned*)(ws + WS_XKEY);
  unsigned* vbm     = (unsigned*)(ws + WS_VBM);
  int*      vrank   = ws + WS_VRANK;
  int*      minxy   = ws + WS_MIN;
  int*      nuniq   = ws + WS_NUNIQ;
  int*      maxabs  = ws + WS_MAXABS;
  int*      bsum    = ws + WS_BSUM;
  int*      boff    = ws + WS_BOFF;
  unsigned* ebm     = (unsigned*)(ws + WS_EBM);

  float* out    = (float*)d_out;
  float* o_x    = out + O_X;
  float* o_pos  = out + O_POS;
  float* o_bat  = out + O_BATCH;
  float* o_ei   = out + O_EI;
  float* o_ea   = out + O_EA;

  k_init_small<<<2048, 256, 0, stream>>>(cnt, first, psum, xkeys, vbm, minxy, nuniq, maxabs);
  k_zero_ebm<<<4096, 256, 0, stream>>>((v4u*)ebm);
  k_minpos<<<1024, 256, 0, stream>>>(pos, minxy);
  k_vox<<<1024, 256, 0, stream>>>(pos, batch, minxy, vox, vbm);
  k_rank<<<1, 1024, 0, stream>>>(vbm, vrank);
  k_pool<<<2048, 256, 0, stream>>>(x, pos, vox, vrank, inv, cnt, first, psum, xkeys);
  k_fin_x<<<(NMAX * FF / 4 + 255) / 256, 256, 0, stream>>>((const v4u*)xkeys, cnt, o_x);
  k_fin_pos<<<(NMAX + 255) / 256, 256, 0, stream>>>(cnt, psum, first, batch, o_pos, o_bat);
  k_mark_edges<<<4096, 256, 0, stream>>>(ei0, ei1, inv, ebm);
  k_blockcount<<<NBLK_E, 256, 0, stream>>>(ebm, bsum);
  k_scan<<<1, 1024, 0, stream>>>(bsum, boff, nuniq);
  k_emit<<<NBLK_E, 256, 0, stream>>>(ebm, boff, o_ei);
  k_cart<<<4096, 256, 0, stream>>>(o_pos, o_ei, o_ea, nuniq, maxabs);
  k_fin_edges<<<4096, 256, 0, stream>>>(o_ei, o_ea, nuniq, maxabs);
}